// NeuralCameraModule_68101001445770
// MI455X (gfx1250) — compile-verified
//
#include <hip/hip_runtime.h>
#include <stdint.h>
#include <math.h>

// NeRF hierarchical render: inverse-CDF importance sampling + fine integration.
// TWO rays per wave32 (16 lanes/ray, 4 samples/lane) to halve per-ray
// instruction count: this kernel is issue-bound (~770 dyn instr/wave vs a
// ~25us HBM floor), so cross-lane stages (scans / butterflies / bitonic sort)
// are amortized over 2 rays and use width-16 shuffles.
// CDNA5 paths: global_load_async_to_lds_b128 staging of color_fine,
// s_wait_asynccnt, LDS tables for searchsorted, wave32 permute network.

constexpr int S = 64;                 // samples (coarse == fine)
constexpr int RAYS_PER_BLOCK = 16;    // 256 threads, 16 lanes per ray
constexpr int BLOCK = 256;

__global__ __launch_bounds__(BLOCK)
void nerf_render_kernel(const float* __restrict__ density_coarse, // [R,64]
                        const float* __restrict__ z_vals,         // [R,64]
                        const float* __restrict__ rays_d,         // [R,3]
                        const float* __restrict__ u_in,           // [R,64]
                        const float* __restrict__ density_fine,   // [R,64]
                        const float* __restrict__ color_fine,     // [R,64,3]
                        float* __restrict__ out,                  // rgb|acc|depth|z_fine
                        int n_rays)
{
    __shared__ __align__(16) float s_color[RAYS_PER_BLOCK][S * 3]; // 12 KB
    __shared__ __align__(16) float s_cdf [RAYS_PER_BLOCK][S];      // 4 KB
    __shared__ __align__(16) float s_bins[RAYS_PER_BLOCK][S];      // 4 KB

    const int l   = threadIdx.x & 15;        // lane within 16-lane ray group
    const int ri  = threadIdx.x >> 4;        // ray slot in block, 0..15
    const int ray = blockIdx.x * RAYS_PER_BLOCK + ri;
    if (ray >= n_rays) return;

    // ---- Async-stage this ray's colors (768B = 48 x 16B, 16 lanes x 3) ----
    {
        const uint64_t g = (uint64_t)(uintptr_t)color_fine
                         + (uint64_t)ray * (S * 3 * 4) + (uint64_t)(l * 16);
        const uint32_t ld = (uint32_t)(uintptr_t)(&s_color[ri][0]) + (uint32_t)(l * 16);
        asm volatile("global_load_async_to_lds_b128 %0, %1, off" :: "v"(ld),        "v"(g)        : "memory");
        asm volatile("global_load_async_to_lds_b128 %0, %1, off" :: "v"(ld + 256u), "v"(g + 256u) : "memory");
        asm volatile("global_load_async_to_lds_b128 %0, %1, off" :: "v"(ld + 512u), "v"(g + 512u) : "memory");
    }
    // Warm GL2 for the fine-density stream while Phase A computes.
    __builtin_prefetch(density_fine + (size_t)ray * S + 4 * l, 0, 0);

    // ---- Ray direction norm (uniform per 16-lane group) ----
    const float* rd = rays_d + (size_t)ray * 3;
    const float nrm = sqrtf(rd[0] * rd[0] + rd[1] * rd[1] + rd[2] * rd[2]);

    // ---- Phase A: coarse transmittance -> weights -> cdf/bins ----
    const float4 z  = ((const float4*)(z_vals         + (size_t)ray * S))[l];
    const float4 dc = ((const float4*)(density_coarse + (size_t)ray * S))[l];
    const float  zn = __shfl_down(z.x, 1, 16);            // z[4l+4], valid l<15

    const float d0 = (z.y - z.x) * 100.0f * nrm;
    const float d1 = (z.z - z.y) * 100.0f * nrm;
    const float d2 = (z.w - z.z) * 100.0f * nrm;
    const float d3 = (l < 15) ? (zn - z.w) * 100.0f * nrm : 1e10f * nrm;
    const float a0 = 1.0f - __expf(-fmaxf(dc.x, 0.0f) * d0);
    const float a1 = 1.0f - __expf(-fmaxf(dc.y, 0.0f) * d1);
    const float a2 = 1.0f - __expf(-fmaxf(dc.z, 0.0f) * d2);
    const float a3 = 1.0f - __expf(-fmaxf(dc.w, 0.0f) * d3);
    const float t0 = 1.0f - a0 + 1e-10f;
    const float t1 = 1.0f - a1 + 1e-10f;
    const float t2 = 1.0f - a2 + 1e-10f;
    const float t3 = 1.0f - a3 + 1e-10f;

    // Exclusive cumprod over 64 (serial-4 per lane + width-16 Kogge-Stone)
    const float c01 = t0 * t1, c012 = c01 * t2, c0123 = c012 * t3;
    float s = c0123;
    #pragma unroll
    for (int d = 1; d < 16; d <<= 1) {
        float o = __shfl_up(s, d, 16);
        if (l >= d) s *= o;
    }
    float e = __shfl_up(s, 1, 16);
    if (l == 0) e = 1.0f;
    const float w0 = a0 * e;
    const float w1 = a1 * (e * t0);
    const float w2 = a2 * (e * c01);
    const float w3 = a3 * (e * c012);

    // pdf weights: coarse-weight indices 1..62, +1e-5; i = 4l+r
    const float wp0 = (l >= 1) ? (w0 + 1e-5f) : 0.0f;     // i=4l   (exclude i=0)
    const float wp1 = w1 + 1e-5f;                          // i=4l+1 always in [1,61]
    const float wp2 = w2 + 1e-5f;                          // i=4l+2 always in [2,62]
    const float wp3 = (l < 15) ? (w3 + 1e-5f) : 0.0f;     // i=4l+3 (exclude i=63)
    const float q1 = wp0 + wp1, q2 = q1 + wp2, q3 = q2 + wp3;
    float ssum = q3;
    #pragma unroll
    for (int d = 1; d < 16; d <<= 1) {
        float o = __shfl_up(ssum, d, 16);
        if (l >= d) ssum += o;
    }
    const float tot = __shfl(ssum, 15, 16);
    float es = __shfl_up(ssum, 1, 16);
    if (l == 0) es = 0.0f;
    const float inv = 1.0f / tot;

    ((float4*)&s_cdf[ri][0])[l] =
        make_float4((es + wp0) * inv, (es + q1) * inv, (es + q2) * inv, (es + q3) * inv);
    ((float4*)&s_bins[ri][0])[l] =
        make_float4(0.5f * (z.x + z.y), 0.5f * (z.y + z.z),
                    0.5f * (z.z + z.w), 0.5f * (z.w + zn));   // index 63 never read

    // ---- Inverse-CDF sampling: 4 u's per lane, 6-step branch-free search ----
    const float4 uu = ((const float4*)(u_in + (size_t)ray * S))[l];
    const float* cdfA = &s_cdf[ri][0];
    const float* binA = &s_bins[ri][0];
    auto sample_z = [&](float uv) {
        int lo = 0, hi = 63;                 // searchsorted right over 63 entries
        #pragma unroll
        for (int it = 0; it < 6; ++it) {     // 63->31->15->7->3->1->0
            const int  mid = (lo + hi) >> 1;
            const bool le  = (cdfA[mid] <= uv);
            lo = le ? (mid + 1) : lo;
            hi = le ? hi        : mid;
        }
        int below = lo - 1; if (below < 0)  below = 0;
        int above = lo;     if (above > 62) above = 62;
        const float c0 = cdfA[below], c1 = cdfA[above];
        const float b0 = binA[below], b1 = binA[above];
        float den = c1 - c0;
        den = (den < 1e-5f) ? 1.0f : den;
        return b0 + (uv - c0) / den * (b1 - b0);
    };
    float v0 = sample_z(uu.x);
    float v1 = sample_z(uu.y);
    float v2 = sample_z(uu.z);
    float v3 = sample_z(uu.w);

    // ---- Bitonic sort of 64 (4 regs/lane, element index i = 4l + r) ----
    auto CE = [](float& a, float& b, bool up) {          // a<=b if up, else a>=b
        const float mn = fminf(a, b), mx = fmaxf(a, b);
        a = up ? mn : mx;
        b = up ? mx : mn;
    };
    CE(v0, v1, true);                                    // stage k=2 (dir from i&2)
    CE(v2, v3, false);
    {                                                    // stage k=4
        const bool asc = ((l & 1) == 0);
        CE(v0, v2, asc); CE(v1, v3, asc);                // j=2
        CE(v0, v1, asc); CE(v2, v3, asc);                // j=1
    }
    #pragma unroll
    for (int k = 8; k <= 64; k <<= 1) {                  // stages k=8..64
        const bool asc = ((l & (k >> 2)) == 0);          // k=64 -> always ascending
        #pragma unroll
        for (int j = k >> 1; j >= 4; j >>= 1) {          // inter-lane steps
            const int  h  = j >> 2;                      // partner lane = l ^ h
            const bool tm = (((l & h) == 0) == asc);
            const float p0 = __shfl_xor(v0, h, 16);
            const float p1 = __shfl_xor(v1, h, 16);
            const float p2 = __shfl_xor(v2, h, 16);
            const float p3 = __shfl_xor(v3, h, 16);
            v0 = tm ? fminf(v0, p0) : fmaxf(v0, p0);
            v1 = tm ? fminf(v1, p1) : fmaxf(v1, p1);
            v2 = tm ? fminf(v2, p2) : fmaxf(v2, p2);
            v3 = tm ? fminf(v3, p3) : fmaxf(v3, p3);
        }
        CE(v0, v2, asc); CE(v1, v3, asc);                // j=2 (register-local)
        CE(v0, v1, asc); CE(v2, v3, asc);                // j=1 (register-local)
    }

    const size_t R = (size_t)n_rays;
    ((float4*)(out + 5 * R + (size_t)ray * S))[l] = make_float4(v0, v1, v2, v3);

    // ---- Phase B: fine integration with opaque far pad ----
    const float4 df  = ((const float4*)(density_fine + (size_t)ray * S))[l];
    const float  znf = __shfl_down(v0, 1, 16);           // z_fine[4l+4]
    const float fd0 = (v1 - v0) * 100.0f * nrm;
    const float fd1 = (v2 - v1) * 100.0f * nrm;
    const float fd2 = (v3 - v2) * 100.0f * nrm;
    const float fd3 = ((l < 15) ? (znf - v3) * 100.0f : (1e8f - v3) * 100.0f) * nrm;
    const float fa0 = 1.0f - __expf(-fmaxf(df.x, 0.0f) * fd0);
    const float fa1 = 1.0f - __expf(-fmaxf(df.y, 0.0f) * fd1);
    const float fa2 = 1.0f - __expf(-fmaxf(df.z, 0.0f) * fd2);
    const float fa3 = 1.0f - __expf(-fmaxf(df.w, 0.0f) * fd3);
    const float ft0 = 1.0f - fa0 + 1e-10f;
    const float ft1 = 1.0f - fa1 + 1e-10f;
    const float ft2 = 1.0f - fa2 + 1e-10f;
    const float ft3 = 1.0f - fa3 + 1e-10f;

    const float fc01 = ft0 * ft1, fc012 = fc01 * ft2, fc0123 = fc012 * ft3;
    float fs = fc0123;
    #pragma unroll
    for (int d = 1; d < 16; d <<= 1) {
        float o = __shfl_up(fs, d, 16);
        if (l >= d) fs *= o;
    }
    const float T = __shfl(fs, 15, 16);                  // pad transmittance
    float fe = __shfl_up(fs, 1, 16);
    if (l == 0) fe = 1.0f;
    const float fw0 = fa0 * fe;
    const float fw1 = fa1 * (fe * ft0);
    const float fw2 = fa2 * (fe * fc01);
    const float fw3 = fa3 * (fe * fc012);

    // Colors staged by the async DMA: wait on this wave's ASYNCcnt only.
    asm volatile("s_wait_asynccnt 0x0" ::: "memory");
    const float4* sc4 = (const float4*)&s_color[ri][0];
    const float4 f0 = sc4[3 * l + 0];   // {r0 g0 b0 r1}
    const float4 f1 = sc4[3 * l + 1];   // {g1 b1 r2 g2}
    const float4 f2 = sc4[3 * l + 2];   // {b2 r3 g3 b3}

    float accL = fw0 + fw1 + fw2 + fw3;
    float depL = fw0 * v0 + fw1 * v1 + fw2 * v2 + fw3 * v3;
    float rL = fw0 * f0.x + fw1 * f0.w + fw2 * f1.z + fw3 * f2.y;
    float gL = fw0 * f0.y + fw1 * f1.x + fw2 * f1.w + fw3 * f2.z;
    float bL = fw0 * f0.z + fw1 * f1.y + fw2 * f2.x + fw3 * f2.w;
    auto wred = [&](float x) {
        #pragma unroll
        for (int d = 8; d >= 1; d >>= 1) x += __shfl_xor(x, d, 16);
        return x;
    };
    accL = wred(accL); depL = wred(depL);
    rL = wred(rL); gL = wred(gL); bL = wred(bL);

    if (l == 0) {
        // pad sample: density 1e8, dist 1e10*nrm, color (1,1,1), z 1e8
        const float apad = 1.0f - __expf(-1e18f * nrm);
        const float wpad = T * apad;
        out[(size_t)ray * 3 + 0] = rL + wpad;
        out[(size_t)ray * 3 + 1] = gL + wpad;
        out[(size_t)ray * 3 + 2] = bL + wpad;
        out[3 * R + ray] = accL + wpad;
        out[4 * R + ray] = depL + wpad * 1e8f;
    }
}

extern "C" void kernel_launch(void* const* d_in, const int* in_sizes, int n_in,
                              void* d_out, int out_size, void* d_ws, size_t ws_size,
                              hipStream_t stream) {
    const float* density_coarse = (const float*)d_in[0];
    const float* z_vals         = (const float*)d_in[1];
    const float* rays_d         = (const float*)d_in[2];
    const float* u              = (const float*)d_in[3];
    const float* density_fine   = (const float*)d_in[4];
    const float* color_fine     = (const float*)d_in[5];
    const int n_rays = in_sizes[0] / S;
    const int blocks = (n_rays + RAYS_PER_BLOCK - 1) / RAYS_PER_BLOCK;
    nerf_render_kernel<<<blocks, BLOCK, 0, stream>>>(
        density_coarse, z_vals, rays_d, u, density_fine, color_fine,
        (float*)d_out, n_rays);
}